// ROIEmbedding_23235773071813
// MI455X (gfx1250) — compile-verified
//
#include <hip/hip_runtime.h>
#include <hip/hip_bf16.h>

// SPP max-pool pyramid for x[32,64,64,512] f32 -> out[32, 85*512].
// Pure streaming max-reduction (256 MiB read @ 23.3 TB/s ~ 11.5 us).
// CDNA5 path: async global->LDS copies (ASYNCcnt) with NT hints + ds_load_b128.
// This revision uses the async instruction's 24-bit IOFFSET (applied to BOTH
// the LDS dest and the global address, per CDNA5 ISA) so a
// whole 16 KiB chunk is issued from one LDS-base VGPR + one global-base pair.

typedef __attribute__((ext_vector_type(4))) float v4f;

__device__ __forceinline__ v4f vmax4(v4f a, v4f b) {
    v4f r;
    r.x = fmaxf(a.x, b.x);
    r.y = fmaxf(a.y, b.y);
    r.z = fmaxf(a.z, b.z);
    r.w = fmaxf(a.w, b.w);
    return r;
}

#define NEG_HUGE (-3.402823466e+38f)

// ---------------------------------------------------------------------------
// Kernel A: finest 8x8 bin maxima.
// Grid: 32*8*8 = 2048 blocks, 128 threads each. Block (b,i8,j8) reduces an
// 8x8-pixel bin over 512 channels. One chunk = one pixel row of the bin
// = 8 pixels * 512 ch * 4B = 16 KiB contiguous. Double-buffered in 2x16 KiB
// dynamic LDS via global_load_async_to_lds_b128 with offset immediates.
// Lane t async-loads float4 indices {k*128+t} and later consumes exactly
// pixel-p float4 index {p*128+t} -> producer lane == consumer lane, so the
// per-wave s_wait_asynccnt is a complete ordering guarantee (no barrier).
// ---------------------------------------------------------------------------
__global__ __launch_bounds__(128) void spp_finest(const float* __restrict__ x,
                                                  float* __restrict__ ws) {
    const int bid  = blockIdx.x;
    const int j8   = bid & 7;
    const int i8   = (bid >> 3) & 7;
    const int b    = bid >> 6;
    const int t    = threadIdx.x;

    // Base of this bin: x[b, i8*8, j8*8, 0]; one h-row chunk is 16 KiB contiguous.
    const size_t rowStride = (size_t)64 * 512;  // floats per h step
    const float* base = x + (((size_t)(b * 64 + i8 * 8)) * 64 + (size_t)(j8 * 8)) * 512;

    // Issue one 16 KiB chunk (h-row) into LDS buffer `buf` (0/1).
    // IOFFSET advances the LDS dest and global src in lockstep, so one base
    // address per side covers the whole chunk.
    auto issue = [&](int h, int buf) {
        const v4f* gp   = (const v4f*)(base + (size_t)h * rowStride) + t;
        const unsigned lds = (unsigned)(buf * 16384 + t * 16);
        asm volatile(
            "global_load_async_to_lds_b128 %0, %1, off th:TH_LOAD_NT\n\t"
            "global_load_async_to_lds_b128 %0, %1, off offset:2048 th:TH_LOAD_NT\n\t"
            "global_load_async_to_lds_b128 %0, %1, off offset:4096 th:TH_LOAD_NT\n\t"
            "global_load_async_to_lds_b128 %0, %1, off offset:6144 th:TH_LOAD_NT\n\t"
            "global_load_async_to_lds_b128 %0, %1, off offset:8192 th:TH_LOAD_NT\n\t"
            "global_load_async_to_lds_b128 %0, %1, off offset:10240 th:TH_LOAD_NT\n\t"
            "global_load_async_to_lds_b128 %0, %1, off offset:12288 th:TH_LOAD_NT\n\t"
            "global_load_async_to_lds_b128 %0, %1, off offset:14336 th:TH_LOAD_NT"
            :: "v"(lds), "v"(gp)
            : "memory");
    };

    v4f acc = {NEG_HUGE, NEG_HUGE, NEG_HUGE, NEG_HUGE};

    issue(0, 0);
#pragma unroll
    for (int h = 0; h < 8; ++h) {
        const int buf = h & 1;
        if (h < 7) {
            issue(h + 1, (h + 1) & 1);
            // chunks h and h+1 in flight (<=16); <=8 left  => chunk h complete
            asm volatile("s_wait_asynccnt 0x8" ::: "memory");
        } else {
            asm volatile("s_wait_asynccnt 0x0" ::: "memory");
        }

        // Consume chunk: 8 pixels, this lane's 4 channels live at
        // byte offset p*2048 + t*16 within the buffer.
        const unsigned a0 = (unsigned)(buf * 16384 + t * 16);
        v4f p0, p1, p2, p3, p4, p5, p6, p7;
        asm volatile(
            "ds_load_b128 %0, %8 offset:0\n\t"
            "ds_load_b128 %1, %8 offset:2048\n\t"
            "ds_load_b128 %2, %8 offset:4096\n\t"
            "ds_load_b128 %3, %8 offset:6144\n\t"
            "ds_load_b128 %4, %8 offset:8192\n\t"
            "ds_load_b128 %5, %8 offset:10240\n\t"
            "ds_load_b128 %6, %8 offset:12288\n\t"
            "ds_load_b128 %7, %8 offset:14336\n\t"
            "s_wait_dscnt 0x0"
            : "=v"(p0), "=v"(p1), "=v"(p2), "=v"(p3),
              "=v"(p4), "=v"(p5), "=v"(p6), "=v"(p7)
            : "v"(a0)
            : "memory");

        v4f m01 = vmax4(p0, p1);
        v4f m23 = vmax4(p2, p3);
        v4f m45 = vmax4(p4, p5);
        v4f m67 = vmax4(p6, p7);
        acc = vmax4(acc, vmax4(vmax4(m01, m23), vmax4(m45, m67)));
    }

    // ws[b, i8, j8, c] as float4: index ((b*8+i8)*8+j8)*128 + t
    v4f* w4 = (v4f*)ws;
    w4[(size_t)bid * 128 + t] = acc;
}

// ---------------------------------------------------------------------------
// Kernel B: hierarchical combine of the 8x8 finest grid into all 4 scales,
// written in the reference concat layout:
//   per-batch float4 offsets (x128): 1x1 @0, 2x2 @1.., 4x4 @5.., 8x8 @21..
// Grid: 32 blocks x 128 threads; thread t owns channels 4t..4t+3.
// ---------------------------------------------------------------------------
__global__ __launch_bounds__(128) void spp_combine(const float* __restrict__ ws,
                                                   float* __restrict__ out) {
    const int b = blockIdx.x;
    const int t = threadIdx.x;

    const v4f* w4 = (const v4f*)(ws + (size_t)b * 64 * 512);
    v4f*       o  = (v4f*)(out + (size_t)b * 85 * 512);

    v4f m2[4];
#pragma unroll
    for (int q = 0; q < 4; ++q) m2[q] = (v4f){NEG_HUGE, NEG_HUGE, NEG_HUGE, NEG_HUGE};

#pragma unroll
    for (int i4 = 0; i4 < 4; ++i4) {
#pragma unroll
        for (int j4 = 0; j4 < 4; ++j4) {
            v4f m4 = {NEG_HUGE, NEG_HUGE, NEG_HUGE, NEG_HUGE};
#pragma unroll
            for (int di = 0; di < 2; ++di) {
#pragma unroll
                for (int dj = 0; dj < 2; ++dj) {
                    const int i8 = i4 * 2 + di;
                    const int j8 = j4 * 2 + dj;
                    v4f v = w4[(size_t)(i8 * 8 + j8) * 128 + t];
                    o[(size_t)(21 + i8 * 8 + j8) * 128 + t] = v;  // 8x8 scale
                    m4 = vmax4(m4, v);
                }
            }
            o[(size_t)(5 + i4 * 4 + j4) * 128 + t] = m4;          // 4x4 scale
            const int q = (i4 >> 1) * 2 + (j4 >> 1);
            m2[q] = vmax4(m2[q], m4);
        }
    }

    v4f m1 = {NEG_HUGE, NEG_HUGE, NEG_HUGE, NEG_HUGE};
#pragma unroll
    for (int q = 0; q < 4; ++q) {
        o[(size_t)(1 + q) * 128 + t] = m2[q];                     // 2x2 scale
        m1 = vmax4(m1, m2[q]);
    }
    o[t] = m1;                                                    // 1x1 scale
}

extern "C" void kernel_launch(void* const* d_in, const int* in_sizes, int n_in,
                              void* d_out, int out_size, void* d_ws, size_t ws_size,
                              hipStream_t stream) {
    (void)in_sizes; (void)n_in; (void)out_size; (void)ws_size;

    const float* x   = (const float*)d_in[0];
    float*       out = (float*)d_out;
    float*       ws  = (float*)d_ws;   // needs 32*8*8*512*4 = 4 MiB

    // Kernel A: 32 batches * 64 finest bins; 32 KiB dynamic LDS (double buffer).
    spp_finest<<<dim3(32 * 8 * 8), dim3(128), 32768, stream>>>(x, ws);
    // Kernel B: combine + write all scales.
    spp_combine<<<dim3(32), dim3(128), 0, stream>>>(ws, out);
}